// EfficientGlobalPointer_72584947302406
// MI455X (gfx1250) — compile-verified
//
#include <hip/hip_runtime.h>
#include <hip/hip_bf16.h>
#include <math.h>

typedef __attribute__((ext_vector_type(2))) float v2f;
typedef __attribute__((ext_vector_type(8))) float v8f;

#define HIDDEN    1024
#define HEADS     12
#define HEAD_SIZE 64
#define NB        16
#define LSEQ      512
#define NROWS     (NB * LSEQ)   // 8192 rows of (B, L)

// ---------------------------------------------------------------------------
// Phase 1: seq_out = inputs @ Wp + bp
//   M = 8192, N = 128, K = 1024.  One wave computes one 16x16 tile using
//   v_wmma_f32_16x16x4_f32 over 256 K-steps.
// A fragment (16x4 f32): lane holds A[m][kb+2*hl], A[m][kb+2*hl+1]  (b64 load)
// B fragment (4x16 f32): lane holds B[kb+2*hl][n], B[kb+2*hl+1][n]
// C/D (16x16 f32): VGPR v <-> row v+8*hl, col lane&15
// ---------------------------------------------------------------------------
__global__ void k_seqout(const float* __restrict__ in, const float* __restrict__ Wp,
                         const float* __restrict__ bp, float* __restrict__ seq) {
  const int wave  = (int)((blockIdx.x * blockDim.x + threadIdx.x) >> 5);
  const int lane  = (int)(threadIdx.x & 31);
  const int tileN = wave & 7;    // 128/16 = 8 column tiles
  const int tileM = wave >> 3;   // 8192/16 = 512 row tiles
  const int lo = lane & 15;
  const int hl = lane >> 4;
  const int m = tileM * 16 + lo;          // this lane's A row
  const int n = tileN * 16 + lo;          // this lane's B col

  const float* arow = in + (size_t)m * HIDDEN + 2 * hl;
  const float* bcol = Wp + (size_t)(2 * hl) * 128 + n;

  v8f c = {0.f, 0.f, 0.f, 0.f, 0.f, 0.f, 0.f, 0.f};
  for (int kb = 0; kb < HIDDEN; kb += 4) {
    v2f a = *(const v2f*)(arow + kb);
    v2f b;
    b.x = bcol[(size_t)kb * 128];
    b.y = bcol[(size_t)kb * 128 + 128];
    c = __builtin_amdgcn_wmma_f32_16x16x4_f32(false, a, false, b, (short)0, c,
                                              false, false);
  }

  const float bias = bp[n];
#pragma unroll
  for (int v = 0; v < 8; ++v) {
    const int row = tileM * 16 + v + 8 * hl;
    seq[(size_t)row * 128 + n] = c[v] + bias;
  }
}

// ---------------------------------------------------------------------------
// Phase 2: bias_input = seq_out(raw) @ Wq + bq    (N = 24, trivial FLOPs)
// Must run BEFORE RoPE mutates seq in place (reference uses un-roped seq_out).
// ---------------------------------------------------------------------------
__global__ void k_bias(const float* __restrict__ seq, const float* __restrict__ Wq,
                       const float* __restrict__ bq, float* __restrict__ bias) {
  const int t = (int)(blockIdx.x * blockDim.x + threadIdx.x);
  if (t >= NROWS * 2 * HEADS) return;
  const int row = t / (2 * HEADS);
  const int j   = t - row * (2 * HEADS);
  const float* s = seq + (size_t)row * 128;
  float acc = bq[j];
  for (int k = 0; k < 128; ++k) acc = fmaf(s[k], Wq[(size_t)k * (2 * HEADS) + j], acc);
  bias[t] = acc;
}

// ---------------------------------------------------------------------------
// Phase 3: in-place RoPE on both q (cols 0..63) and k (cols 64..127) halves.
//   out[2i]   = x[2i]*cos_i - x[2i+1]*sin_i
//   out[2i+1] = x[2i+1]*cos_i + x[2i]*sin_i,  theta_i = pos * 10000^(-i/32)
// ---------------------------------------------------------------------------
__global__ void k_rope(float* __restrict__ seq) {
  const int t = (int)(blockIdx.x * blockDim.x + threadIdx.x);
  if (t >= NROWS * 64) return;                 // 2 halves * 32 pairs per row
  const int row  = t >> 6;
  const int r    = t & 63;
  const int half = r >> 5;
  const int i    = r & 31;
  const int pos  = row & (LSEQ - 1);
  // freq = exp(-2i * ln(10000)/64)
  const float freq = expf(-(float)i * 0.2878231366f);
  const float ang  = (float)pos * freq;
  float s, c;
  sincosf(ang, &s, &c);
  float* p = seq + (size_t)row * 128 + half * 64 + 2 * i;
  const float x0 = p[0], x1 = p[1];
  p[0] = x0 * c - x1 * s;
  p[1] = x1 * c + x0 * s;
}

// ---------------------------------------------------------------------------
// Phase 4: logits. One wave per 16x16 tile of qk per batch.
//   qk = qw @ kw^T / 8     (K = 64 -> 16 WMMAs)
//   out[b,h,r,c] = qk[r,c] + biasQ[b,r,h] + biasK[b,c,h] - (r>c)*1e12,
//   masked rows/cols -> -inf.  B fragment B[k][n] = kw[n][k]: contiguous load.
//   12 head copies written with nontemporal stores (201 MB total; keep L2 for
//   the 4 MB seq buffer which is re-read 32x).
// ---------------------------------------------------------------------------
__global__ void k_logits(const float* __restrict__ seq, const float* __restrict__ bias,
                         const int* __restrict__ mask, float* __restrict__ out) {
  const int wave = (int)((blockIdx.x * blockDim.x + threadIdx.x) >> 5);
  const int lane = (int)(threadIdx.x & 31);
  const int tj = wave & 31;          // 512/16 = 32 col tiles
  const int ti = (wave >> 5) & 31;   // 32 row tiles
  const int b  = wave >> 10;         // batch
  const int lo = lane & 15;
  const int hl = lane >> 4;
  const int iBase = ti * 16, jBase = tj * 16;

  const float* aptr = seq + (size_t)(b * LSEQ + iBase + lo) * 128 + 2 * hl;       // qw
  const float* bptr = seq + (size_t)(b * LSEQ + jBase + lo) * 128 + 64 + 2 * hl;  // kw

  v8f c = {0.f, 0.f, 0.f, 0.f, 0.f, 0.f, 0.f, 0.f};
#pragma unroll
  for (int kb = 0; kb < HEAD_SIZE; kb += 4) {
    v2f a  = *(const v2f*)(aptr + kb);
    v2f bf = *(const v2f*)(bptr + kb);
    c = __builtin_amdgcn_wmma_f32_16x16x4_f32(false, a, false, bf, (short)0, c,
                                              false, false);
  }

  const float scale = 0.125f;  // 1/sqrt(64)
  const int   col   = jBase + lo;
  const int   colMask = mask[b * LSEQ + col];

  float base[8];
  int   rowIdx[8];
  int   rowMask[8];
#pragma unroll
  for (int v = 0; v < 8; ++v) {
    const int row = iBase + v + 8 * hl;
    rowIdx[v]  = row;
    rowMask[v] = mask[b * LSEQ + row];
    float val = c[v] * scale;
    if (row > col) val -= 1.0e12f;   // strict lower triangle
    base[v] = val;
  }

  const float* biasCol = bias + (size_t)(b * LSEQ + col) * (2 * HEADS);
#pragma unroll
  for (int h = 0; h < HEADS; ++h) {
    const float ck = biasCol[2 * h + 1];
    const size_t headBase = ((size_t)(b * HEADS + h)) * LSEQ * LSEQ;
#pragma unroll
    for (int v = 0; v < 8; ++v) {
      const float rq = bias[(size_t)(b * LSEQ + rowIdx[v]) * (2 * HEADS) + 2 * h];
      float val = base[v] + rq + ck;
      if (!rowMask[v] || !colMask) val = -__builtin_inff();
      __builtin_nontemporal_store(val, out + headBase + (size_t)rowIdx[v] * LSEQ + col);
    }
  }
}

// ---------------------------------------------------------------------------
extern "C" void kernel_launch(void* const* d_in, const int* in_sizes, int n_in,
                              void* d_out, int out_size, void* d_ws, size_t ws_size,
                              hipStream_t stream) {
  (void)in_sizes; (void)n_in; (void)out_size; (void)ws_size;
  const float* inputs = (const float*)d_in[0];
  // d_in[1] = event_inputs: unused by the reference output
  const int*   mask = (const int*)d_in[2];
  const float* Wp = (const float*)d_in[3];
  const float* bp = (const float*)d_in[4];
  const float* Wq = (const float*)d_in[5];
  const float* bq = (const float*)d_in[6];

  float* seq  = (float*)d_ws;                        // 8192*128 f32 = 4 MiB
  float* bias = seq + (size_t)NROWS * 128;           // 8192*24 f32 = 768 KiB
  float* out  = (float*)d_out;

  // 4096 tile-waves, 8 waves (256 thr) per block
  k_seqout<<<512, 256, 0, stream>>>(inputs, Wp, bp, seq);
  // 8192*24 threads
  k_bias<<<(NROWS * 2 * HEADS + 255) / 256, 256, 0, stream>>>(seq, Wq, bq, bias);
  // 8192*64 pair-threads
  k_rope<<<(NROWS * 64 + 255) / 256, 256, 0, stream>>>(seq);
  // 16*32*32 = 16384 tile-waves, 8 waves per block
  k_logits<<<2048, 256, 0, stream>>>(seq, bias, mask, out);
}